// LenardJones_20650202759880
// MI455X (gfx1250) — compile-verified
//
#include <hip/hip_runtime.h>
#include <hip/hip_bf16.h>

#define N_PART 4096
#define TILE 16
#define NTILES (N_PART / TILE)            // 256 tiles of 16 rows
#define WAVES_PER_BLOCK 4
#define BLOCK_THREADS (WAVES_PER_BLOCK * 32)
#define NBLOCKS (NTILES / WAVES_PER_BLOCK) // 64 blocks -> 256 waves, one i-tile each

typedef __attribute__((ext_vector_type(2))) float v2f;
typedef __attribute__((ext_vector_type(8))) float v8f;

// dq = p / m  (12288 elements)
__global__ void lj_dq_kernel(const float* __restrict__ p, const float* __restrict__ m,
                             float* __restrict__ out) {
  int idx = blockIdx.x * blockDim.x + threadIdx.x;
  if (idx < 3 * N_PART) out[idx] = p[idx] / m[idx / 3];
}

// dp[i] = -48 * sum_j (r^-14 - 0.5 r^-8) (q_i - q_j)
//       = -48 * ( q_i * S_i - C_i ),  S_i = sum_j c_ij,  C_i = sum_j c_ij q_j
// r2 via Gram matrix on the FP32 matrix core: r2 = |qi|^2 + |qj|^2 - 2 qi.qj
__global__ __launch_bounds__(BLOCK_THREADS)
void lj_force_kernel(const float* __restrict__ q, float* __restrict__ out) {
  __shared__ float lx[N_PART], ly[N_PART], lz[N_PART], ls[N_PART]; // 64 KB

  const int tid = threadIdx.x;
  // Stage all of q (48 KB) + |q|^2 into LDS once; inner O(N^2) loop never hits HBM.
  for (int k = tid; k < N_PART; k += BLOCK_THREADS) {
    float x = q[3 * k + 0], y = q[3 * k + 1], z = q[3 * k + 2];
    lx[k] = x; ly[k] = y; lz[k] = z;
    ls[k] = x * x + y * y + z * z;
  }
  __syncthreads();

  const int lane = tid & 31;
  const int wave = tid >> 5;
  const int ib   = (blockIdx.x * WAVES_PER_BLOCK + wave) * TILE; // i-tile base row
  const int n    = lane & 15;       // column / A-row index within tile
  const bool lo  = lane < 16;       // half-wave selector
  const int hi   = lane >> 4;

  // A operand: 16x4 fp32. lanes 0-15: V0=K0(x), V1=K1(y); lanes 16-31: V0=K2(z), V1=K3(0)
  v2f a;
  a.x = lo ? lx[ib + n] : lz[ib + n];
  a.y = lo ? ly[ib + n] : 0.0f;

  // Rows this lane owns in the C/D layout: row = v + 8*hi (v = VGPR index 0..7)
  int   rowg[8];
  float sim[8];
#pragma unroll
  for (int v = 0; v < 8; ++v) { rowg[v] = ib + v + 8 * hi; sim[v] = ls[rowg[v]]; }

  float accS[8], accX[8], accY[8], accZ[8];
#pragma unroll
  for (int v = 0; v < 8; ++v) { accS[v] = 0.f; accX[v] = 0.f; accY[v] = 0.f; accZ[v] = 0.f; }

  for (int jt = 0; jt < NTILES; ++jt) {
    const int j = jt * TILE + n;
    const float xj = lx[j], yj = ly[j], zj = lz[j], sj = ls[j]; // broadcast ds_loads

    // B operand: 4x16 fp32. lanes 0-15: V0=K0(x), V1=K1(y); lanes 16-31: V0=K2(z), V1=K3(0)
    v2f b;
    b.x = lo ? xj : zj;
    b.y = lo ? yj : 0.0f;

    v8f c = {};
    // g[v] = q_i . q_j for (row = v+8*hi, col = n)
    v8f g = __builtin_amdgcn_wmma_f32_16x16x4_f32(false, a, false, b, (short)0, c,
                                                  false, false);

#pragma unroll
    for (int v = 0; v < 8; ++v) {
      float r2   = __builtin_fmaf(-2.0f, g[v], sim[v] + sj);
      float inv  = __builtin_amdgcn_rcpf(r2);      // v_rcp_f32
      float inv2 = inv * inv;
      float inv4 = inv2 * inv2;
      float inv7 = inv4 * inv2 * inv;
      float cc   = __builtin_fmaf(-0.5f, inv4, inv7); // r^-14 - 0.5 r^-8
      cc = (rowg[v] == j) ? 0.0f : cc;                // kill diagonal (also kills NaN)
      accS[v] += cc;
      accX[v] = __builtin_fmaf(cc, xj, accX[v]);
      accY[v] = __builtin_fmaf(cc, yj, accY[v]);
      accZ[v] = __builtin_fmaf(cc, zj, accZ[v]);
    }
  }

  // Butterfly-reduce the 16 column-lanes of each half-wave (masks < 16 stay in-half).
#pragma unroll
  for (int v = 0; v < 8; ++v) {
#pragma unroll
    for (int msk = 8; msk >= 1; msk >>= 1) {
      accS[v] += __shfl_xor(accS[v], msk, 32);
      accX[v] += __shfl_xor(accX[v], msk, 32);
      accY[v] += __shfl_xor(accY[v], msk, 32);
      accZ[v] += __shfl_xor(accZ[v], msk, 32);
    }
  }

  // Lane with n==v (in each half) writes row v + 8*hi. Deterministic, no atomics.
  float* dp = out + 3 * N_PART;
#pragma unroll
  for (int v = 0; v < 8; ++v) {
    if (n == v) {
      const int row = rowg[v];
      const float S = accS[v];
      dp[3 * row + 0] = -48.0f * (lx[row] * S - accX[v]);
      dp[3 * row + 1] = -48.0f * (ly[row] * S - accY[v]);
      dp[3 * row + 2] = -48.0f * (lz[row] * S - accZ[v]);
    }
  }
}

extern "C" void kernel_launch(void* const* d_in, const int* in_sizes, int n_in,
                              void* d_out, int out_size, void* d_ws, size_t ws_size,
                              hipStream_t stream) {
  (void)in_sizes; (void)n_in; (void)d_ws; (void)ws_size; (void)out_size;
  const float* q = (const float*)d_in[0];
  const float* p = (const float*)d_in[1];
  const float* m = (const float*)d_in[2];
  float* out = (float*)d_out;

  lj_dq_kernel<<<(3 * N_PART + 255) / 256, 256, 0, stream>>>(p, m, out);
  lj_force_kernel<<<NBLOCKS, BLOCK_THREADS, 0, stream>>>(q, out);
}